// DGNN_40922448396353
// MI455X (gfx1250) — compile-verified
//
#include <hip/hip_runtime.h>
#include <hip/hip_bf16.h>

typedef __attribute__((ext_vector_type(2))) float v2f;
typedef __attribute__((ext_vector_type(8))) float v8f;

#define NN    200000   // N_NODES
#define DDIM  128      // message / neighbor dim
#define KNEI  20       // neighbors per node

// ---- order-preserving float <-> uint key (for atomic max over signed floats) ----
__device__ __forceinline__ unsigned enc_f32(float f) {
    unsigned u = __float_as_uint(f);
    return (u & 0x80000000u) ? ~u : (u | 0x80000000u);
}
__device__ __forceinline__ float dec_f32(unsigned u) {
    return (u & 0x80000000u) ? __uint_as_float(u ^ 0x80000000u) : __uint_as_float(~u);
}

// ---- zero the accumulator regions (d_out msg+t region, ws counts/stats) ----
__global__ void k_init(float* __restrict__ oz, long n_out, float* __restrict__ wz, long n_ws) {
    long stride = (long)gridDim.x * blockDim.x;
    long t0 = (long)blockIdx.x * blockDim.x + threadIdx.x;
    for (long i = t0; i < n_out; i += stride) oz[i] = 0.0f;
    for (long i = t0; i < n_ws;  i += stride) wz[i] = 0.0f;
}

// ---- one wave per event: coalesced float4 read, L2-resident f32 atomics ----
__global__ void k_scatter(const int* __restrict__ ids, const float* __restrict__ msg,
                          const float* __restrict__ ts, float* __restrict__ msg_sum,
                          float* __restrict__ counts, unsigned* __restrict__ tmax, int E) {
    int w    = (blockIdx.x * blockDim.x + threadIdx.x) >> 5;
    int lane = threadIdx.x & 31;
    if (w >= E) return;
    int id = ids[w];
    if (lane == 0 && (w + 8) < E)
        __builtin_prefetch(msg + (size_t)(w + 8) * DDIM, 0, 1);   // global_prefetch_b8
    float4 m = *(const float4*)(msg + (size_t)w * DDIM + lane * 4);
    float* dst = msg_sum + (size_t)id * DDIM + lane * 4;
    unsafeAtomicAdd(dst + 0, m.x);
    unsafeAtomicAdd(dst + 1, m.y);
    unsafeAtomicAdd(dst + 2, m.z);
    unsafeAtomicAdd(dst + 3, m.w);
    if (lane == 0) {
        unsafeAtomicAdd(counts + id, 1.0f);
        atomicMax(tmax + id, __float_as_uint(ts[w]));  // ts >= 0: raw bits order correctly
    }
}

// ---- mean = sum / max(count,1); merged_t already final (0 for empty segments) ----
__global__ void k_finalize(float* __restrict__ msg_sum, const float* __restrict__ counts, int n) {
    int t = blockIdx.x * blockDim.x + threadIdx.x;   // n = NN*32 (one float4 each)
    if (t >= n) return;
    int node = t >> 5, d4 = t & 31;
    float inv = 1.0f / fmaxf(counts[node], 1.0f);
    float4* p = (float4*)(msg_sum + (size_t)node * DDIM + d4 * 4);
    float4 v = *p;
    v.x *= inv; v.y *= inv; v.z *= inv; v.w *= inv;
    *p = v;
}

// ---- s2[b,:] = sum_k nei2[b,k,:] ; one thread per float4 of output ----
__global__ void k_s2(const float* __restrict__ nei2, float* __restrict__ s2, int n) {
    int t = blockIdx.x * blockDim.x + threadIdx.x;   // n = B*32
    if (t >= n) return;
    int b = t >> 5, d4 = t & 31;
    const float4* src = (const float4*)(nei2 + (size_t)b * (KNEI * DDIM)) + d4;
    float4 acc = {0.f, 0.f, 0.f, 0.f};
    #pragma unroll
    for (int k = 0; k < KNEI; ++k) {
        float4 v = src[k * (DDIM / 4)];
        acc.x += v.x; acc.y += v.y; acc.z += v.z; acc.w += v.w;
    }
    *((float4*)(s2 + (size_t)b * DDIM) + d4) = acc;
}

// ---- logits[b,k] = nei1[b,k,:] . s2[b,:] via V_WMMA_F32_16X16X4_F32 ----
// One wave per b. A = nei1[b] (rows=k, padded LDS stride 132 -> conflict-free
// column reads: bank stride 4, half-0 banks 4r+c vs half-1 banks 4r'+c+2 are
// parity-disjoint). B operand = s2 chunk replicated across all 16 N columns,
// so every D column equals the wanted dot product; lanes 0/16 hold N=0.
// Tile-1 rows >= 20 read from a shared LDS zero row selected ONCE before the
// loop, so the hot loop is branch-free with EXEC all-ones throughout.
__global__ __launch_bounds__(128) void k_logits(const float* __restrict__ nei1,
                                                const float* __restrict__ s2,
                                                float* __restrict__ logits,
                                                unsigned* __restrict__ colmax, int B) {
    __shared__ float    sA[4][KNEI * 132];
    __shared__ float    sS2[4][DDIM];
    __shared__ float    sZero[DDIM];
    __shared__ unsigned sMax[KNEI];
    const int wave = threadIdx.x >> 5;
    const int lane = threadIdx.x & 31;
    const int b    = blockIdx.x * 4 + wave;
    if (threadIdx.x < KNEI) sMax[threadIdx.x] = 0u;  // encoded -inf
    sZero[threadIdx.x] = 0.0f;                       // blockDim == DDIM == 128

    // cooperative coalesced stage of nei1[b] (20x128) and s2[b] (128) into LDS
    const float4* src = (const float4*)(nei1 + (size_t)b * (KNEI * DDIM));
    for (int i = lane; i < KNEI * (DDIM / 4); i += 32) {
        int k = i >> 5, d4 = i & 31;                  // 32 float4 per row
        float4 v = src[i];
        *(float4*)&sA[wave][k * 132 + d4 * 4] = v;    // 528B row stride, 16B aligned
    }
    {
        float4 v = *((const float4*)(s2 + (size_t)b * DDIM) + lane);
        *(float4*)&sS2[wave][lane * 4] = v;
    }
    __syncthreads();

    // A 16x4 layout: lanes 0-15 -> M=lane, K={c,c+1}; lanes 16-31 -> M=lane-16, K={c+2,c+3}
    const int   half = lane >> 4;
    const int   r    = lane & 15;
    const int   off  = half * 2;
    const float* A0  = &sA[wave][r * 132];                          // M tile 0: rows 0..15
    const float* A1  = (r < 4) ? &sA[wave][(16 + r) * 132] : sZero; // M tile 1: rows 16..19, rest zero
    const float* S   = sS2[wave];

    v8f acc0 = {0.f,0.f,0.f,0.f,0.f,0.f,0.f,0.f};
    v8f acc1 = {0.f,0.f,0.f,0.f,0.f,0.f,0.f,0.f};
    #pragma unroll
    for (int c = 0; c < DDIM; c += 4) {
        v2f bv = { S[c + off],  S[c + off + 1]  };    // replicated-column B operand
        v2f a0 = { A0[c + off], A0[c + off + 1] };
        v2f a1 = { A1[c + off], A1[c + off + 1] };
        acc0 = __builtin_amdgcn_wmma_f32_16x16x4_f32(false, a0, false, bv,
                                                     (short)0, acc0, false, false);
        acc1 = __builtin_amdgcn_wmma_f32_16x16x4_f32(false, a1, false, bv,
                                                     (short)0, acc1, false, false);
    }

    // C/D layout: VGPR v -> M=v (lanes 0-15) / M=v+8 (lanes 16-31); take N=0 column.
    float* lg = logits + (size_t)b * KNEI;
    if (lane == 0) {
        #pragma unroll
        for (int i = 0; i < 8; ++i) { lg[i] = acc0[i]; atomicMax(&sMax[i], enc_f32(acc0[i])); }
        #pragma unroll
        for (int i = 0; i < 4; ++i) { lg[16 + i] = acc1[i]; atomicMax(&sMax[16 + i], enc_f32(acc1[i])); }
    } else if (lane == 16) {
        #pragma unroll
        for (int i = 0; i < 8; ++i) { lg[8 + i] = acc0[i]; atomicMax(&sMax[8 + i], enc_f32(acc0[i])); }
    }
    __syncthreads();
    if (threadIdx.x < KNEI) atomicMax(&colmax[threadIdx.x], sMax[threadIdx.x]);
}

// ---- per-column sum of exp(logit - colmax), LDS-binned then 20 global atomics ----
__global__ void k_sumexp(const float* __restrict__ logits, const unsigned* __restrict__ colmax,
                         float* __restrict__ colsum, int total) {
    __shared__ float sSum[KNEI];
    __shared__ float sMaxv[KNEI];
    if (threadIdx.x < KNEI) { sSum[threadIdx.x] = 0.f; sMaxv[threadIdx.x] = dec_f32(colmax[threadIdx.x]); }
    __syncthreads();
    int stride = gridDim.x * blockDim.x;
    for (int i = blockIdx.x * blockDim.x + threadIdx.x; i < total; i += stride) {
        int k = i % KNEI;
        unsafeAtomicAdd(&sSum[k], expf(logits[i] - sMaxv[k]));
    }
    __syncthreads();
    if (threadIdx.x < KNEI) unsafeAtomicAdd(&colsum[threadIdx.x], sSum[threadIdx.x]);
}

// ---- score = exp(logit - colmax) / colsum ----
__global__ void k_score(const float* __restrict__ logits, const unsigned* __restrict__ colmax,
                        const float* __restrict__ colsum, float* __restrict__ score, int total) {
    int i = blockIdx.x * blockDim.x + threadIdx.x;
    if (i >= total) return;
    int k = i % KNEI;
    score[i] = expf(logits[i] - dec_f32(colmax[k])) / colsum[k];
}

extern "C" void kernel_launch(void* const* d_in, const int* in_sizes, int n_in,
                              void* d_out, int out_size, void* d_ws, size_t ws_size,
                              hipStream_t stream) {
    const int*   node_ids   = (const int*)d_in[0];
    const float* message    = (const float*)d_in[1];
    const float* timestamps = (const float*)d_in[2];
    const float* nei1       = (const float*)d_in[3];
    const float* nei2       = (const float*)d_in[4];
    const int E = in_sizes[0];
    const int B = in_sizes[3] / (KNEI * DDIM);

    // d_out = [merged_message NNxD | merged_t NN | score BxK]
    float* merged_message = (float*)d_out;
    float* merged_t       = merged_message + (size_t)NN * DDIM;
    float* score          = merged_t + NN;

    // workspace layout (floats): counts | colmax | colsum | s2 | logits
    float*    ws     = (float*)d_ws;
    float*    counts = ws;                       // NN
    unsigned* colmax = (unsigned*)(ws + NN);     // 20 (encoded)
    float*    colsum = ws + NN + 32;             // 20
    float*    s2     = ws + 262144;              // B*128
    float*    logits = s2 + (size_t)B * DDIM;    // B*20

    const long n_out_zero = (long)NN * DDIM + NN;         // sums + t accumulators
    const long n_ws_zero  = NN + 64;                      // counts + colmax + colsum
    k_init<<<2048, 256, 0, stream>>>(merged_message, n_out_zero, counts, n_ws_zero);

    k_scatter<<<(E * 32 + 255) / 256, 256, 0, stream>>>(node_ids, message, timestamps,
                                                        merged_message, counts,
                                                        (unsigned*)merged_t, E);

    k_finalize<<<(NN * 32 + 255) / 256, 256, 0, stream>>>(merged_message, counts, NN * 32);

    k_s2<<<(B * 32 + 255) / 256, 256, 0, stream>>>(nei2, s2, B * 32);

    k_logits<<<B / 4, 128, 0, stream>>>(nei1, s2, logits, colmax, B);

    k_sumexp<<<512, 256, 0, stream>>>(logits, colmax, colsum, B * KNEI);

    k_score<<<(B * KNEI + 255) / 256, 256, 0, stream>>>(logits, colmax, colsum, score, B * KNEI);
}